// SplineTransform_73950746903164
// MI455X (gfx1250) — compile-verified
//
#include <hip/hip_runtime.h>
#include <hip/hip_bf16.h>

// SplineTransform, MI455X / gfx1250.
// HBM-bound streaming kernel: 512 MB compulsory traffic -> ~22us floor @ 23.3 TB/s.
// CDNA5 paths used: async global->LDS DMA staging of the coeff LUT (ASYNCcnt),
// bank-padded wave32 LDS gather, non-temporal load/store hints.

#define DIMV            1024
#define GSIZE           16
#define LDS_STRIDE      18        // 16 + 2 pad words: stride-18 => 32 distinct banks across a wave
#define BLOCK           256       // 8 wave32
#define DPT             2         // columns per thread (d = t, t+256) -> 512-col tile per block
#define COLS_PER_BLOCK  (BLOCK * DPT)
#define ROWS_PER_BLOCK  64
#define X_MIN_F         (-3.5f)
#define X_MAX_F         (3.5f)

// Pointer types the async-to-LDS builtin expects (per hipcc diagnostic:
// param 0 is `int __attribute__((vector_size(16)))*` in the global AS).
typedef int v4i __attribute__((vector_size(16)));
typedef v4i __attribute__((address_space(1)))* g_v4i_p;   // global
typedef v4i __attribute__((address_space(3)))* l_v4i_p;   // LDS

__global__ __launch_bounds__(BLOCK)
void spline_fwd_gfx1250(const float* __restrict__ x,
                        const float* __restrict__ coeff,
                        const float* __restrict__ bscale,
                        const float* __restrict__ bbias,
                        const float* __restrict__ grid,
                        float* __restrict__ out,
                        int nrows)
{
    extern __shared__ float smem[];             // COLS_PER_BLOCK * LDS_STRIDE floats (36 KB)

    const int t  = threadIdx.x;
    const int c0 = blockIdx.x * COLS_PER_BLOCK; // column-tile origin

    // ---- Stage coeff[c0 .. c0+511][0..15] into bank-padded LDS via CDNA5 async DMA ----
#pragma unroll
    for (int j = 0; j < DPT; ++j) {
        const int dl = t + j * BLOCK;                       // local column 0..511
        float* src = (float*)(coeff + (size_t)(c0 + dl) * GSIZE);
        float* dst = &smem[dl * LDS_STRIDE];
#if __has_builtin(__builtin_amdgcn_global_load_async_to_lds_b128)
#pragma unroll
        for (int k = 0; k < 4; ++k)
            __builtin_amdgcn_global_load_async_to_lds_b128(
                (g_v4i_p)(src + 4 * k),
                (l_v4i_p)(dst + 4 * k),
                0, 0);
#else
#pragma unroll
        for (int k = 0; k < GSIZE; ++k) dst[k] = src[k];
#endif
    }
#if __has_builtin(__builtin_amdgcn_s_wait_asynccnt)
    __builtin_amdgcn_s_wait_asynccnt(0);
#else
    asm volatile("s_wait_asynccnt 0" ::: "memory");
#endif
    __syncthreads();

    // ---- Per-thread constants (uniform grid: h = (g15-g0)/15) ----
    const float g0      = grid[0];
    const float h       = (grid[GSIZE - 1] - g0) * (1.0f / (float)(GSIZE - 1));
    const float inv_h   = 1.0f / h;
    const float inv_den = 1.0f / (h + 1e-8f);   // reference denominator (x1-x0+1e-8)

    float scl[DPT], bia[DPT];
    int   lb[DPT];
#pragma unroll
    for (int j = 0; j < DPT; ++j) {
        const int d = c0 + t + j * BLOCK;
        scl[j] = bscale[d];
        bia[j] = bbias[d];
        lb[j]  = (t + j * BLOCK) * LDS_STRIDE;
    }

    const int row0 = blockIdx.y * ROWS_PER_BLOCK;
    const int rend = (row0 + ROWS_PER_BLOCK < nrows) ? (row0 + ROWS_PER_BLOCK) : nrows;

#pragma unroll 4
    for (int n = row0; n < rend; ++n) {
        const float* xr = x   + (size_t)n * DIMV + c0 + t;
        float*       yr = out + (size_t)n * DIMV + c0 + t;

        float xv[DPT];
#pragma unroll
        for (int j = 0; j < DPT; ++j)
            xv[j] = __builtin_nontemporal_load(xr + j * BLOCK);   // streamed once: NT

#pragma unroll
        for (int j = 0; j < DPT; ++j) {
            const float xc  = fminf(fmaxf(xv[j], X_MIN_F), X_MAX_F);
            // searchsorted(side='left') on uniform grid:
            // idx = #(grid < xc) ; idx0 = clamp(idx-1,0,14) == clamp(ceil((xc-g0)/h)-1,0,14)
            const float u   = (xc - g0) * inv_h;
            const float i0f = fminf(fmaxf(__builtin_ceilf(u) - 1.0f, 0.0f),
                                    (float)(GSIZE - 2));
            const int   i0  = (int)i0f;
            const float y0  = smem[lb[j] + i0];          // padded LDS: ~conflict-free
            const float y1  = smem[lb[j] + i0 + 1];      // adjacent word (ds_load_2addr-able)
            const float x0  = __builtin_fmaf(i0f, h, g0);
            const float tt  = (xc - x0) * inv_den;
            float r = __builtin_fmaf(scl[j], xc, bia[j]);
            r += __builtin_fmaf(tt, y1 - y0, y0);
            __builtin_nontemporal_store(r, yr + j * BLOCK);       // written once: NT
        }
    }
}

extern "C" void kernel_launch(void* const* d_in, const int* in_sizes, int n_in,
                              void* d_out, int out_size, void* d_ws, size_t ws_size,
                              hipStream_t stream) {
    const float* x      = (const float*)d_in[0];
    const float* coeff  = (const float*)d_in[1];
    const float* bscale = (const float*)d_in[2];
    const float* bbias  = (const float*)d_in[3];
    const float* grid   = (const float*)d_in[4];
    float*       out    = (float*)d_out;

    const int N = in_sizes[0] / DIMV;                         // 65536 rows
    dim3 grd(DIMV / COLS_PER_BLOCK,                           // 2 column tiles
             (N + ROWS_PER_BLOCK - 1) / ROWS_PER_BLOCK);      // 1024 row tiles
    const size_t shmem = (size_t)COLS_PER_BLOCK * LDS_STRIDE * sizeof(float); // 36 KB

    spline_fwd_gfx1250<<<grd, dim3(BLOCK), shmem, stream>>>(
        x, coeff, bscale, bbias, grid, out, N);
}